// VQ_86139864089353
// MI455X (gfx1250) — compile-verified
//
#include <hip/hip_runtime.h>
#include <stdint.h>

// VQ per-dimension nearest-neighbor:  B*D = 786432 queries, K = 512 codewords/dim.
// Strategy: sort codebook per dim (stable), stage sorted tables in LDS via TDM
// (tensor_load_to_lds, TENSORcnt), branchless binary search per query.
// Memory path: b128 vector loads/stores (4 elements/lane) -> ~0.4us HBM floor.

#define KCODES 512
#define DDIMS 3
#define TBL_ELEMS (DDIMS * KCODES * 2)   // 1536 float values + 1536 u32 indices = 3072 dwords = 12KB
#define SORT_THREADS 256
#define MAIN_THREADS 256
#define ITEMS 4                          // consecutive dwords per lane -> b128

typedef unsigned int v4u __attribute__((ext_vector_type(4)));
typedef int          v8i __attribute__((ext_vector_type(8)));
typedef int          v4i __attribute__((ext_vector_type(4)));

// ---------------------------------------------------------------------------
// Kernel 1: stable bitonic sort of each dimension's 512 codewords in LDS.
// Writes: ws[0      .. 6143]  = sorted values,  float  [3][512]
//         ws[6144   .. 12287] = original index, uint32 [3][512]
// ---------------------------------------------------------------------------
__global__ __launch_bounds__(SORT_THREADS)
void vq_sort_codebook(const float* __restrict__ e,
                      float* __restrict__ wsV,
                      unsigned* __restrict__ wsI) {
  __shared__ float    sv[KCODES];
  __shared__ unsigned si[KCODES];
  const int d = blockIdx.x;

  for (int k = threadIdx.x; k < KCODES; k += SORT_THREADS) {
    sv[k] = e[k * DDIMS + d];
    si[k] = (unsigned)k;
  }
  __syncthreads();

  // Bitonic sort; tie-break on original index -> stable -> exact-duplicate
  // codewords sit in ascending k; upper_bound makes the left candidate the largest k of its run.
  for (unsigned size = 2; size <= KCODES; size <<= 1) {
    for (unsigned stride = size >> 1; stride > 0; stride >>= 1) {
      unsigned t  = threadIdx.x;
      unsigned lo = 2u * t - (t & (stride - 1u));
      unsigned hi = lo + stride;
      float    a  = sv[lo], b  = sv[hi];
      unsigned ia = si[lo], ib = si[hi];
      bool asc = ((lo & size) == 0u);
      bool gt  = (a > b) || ((a == b) && (ia > ib));
      bool doSwap = asc ? gt : !gt;
      if (doSwap) {
        sv[lo] = b; sv[hi] = a;
        si[lo] = ib; si[hi] = ia;
      }
      __syncthreads();
    }
  }

  for (int k = threadIdx.x; k < KCODES; k += SORT_THREADS) {
    wsV[d * KCODES + k] = sv[k];
    wsI[d * KCODES + k] = si[k];
  }
}

// ---------------------------------------------------------------------------
// One query: branchless upper_bound over 512 sorted values (9 ds_load steps),
// then resolve between the two bracketing candidates.
// ---------------------------------------------------------------------------
__device__ __forceinline__
void vq_query(const float* __restrict__ sV, const unsigned* __restrict__ sI,
              int d, float x, int& kOut, float& vqOut) {
  const float* Vd = sV + d * KCODES;
  unsigned pos = 0u;
#pragma unroll
  for (unsigned s = KCODES / 2; s > 0; s >>= 1) {
    unsigned np = pos + s;
    float v = Vd[np - 1u];                      // ds_load_b32
    pos = (v <= x) ? np : pos;
  }
  unsigned c0 = (pos == 0u) ? 0u : pos - 1u;
  unsigned c1 = (pos >= (unsigned)KCODES) ? (unsigned)(KCODES - 1) : pos;
  float v0 = Vd[c0];
  float v1 = Vd[c1];
  bool left = fabsf(x - v0) <= fabsf(x - v1);
  unsigned c = left ? c0 : c1;
  vqOut = left ? v0 : v1;
  kOut  = (int)sI[d * KCODES + c];
}

// ---------------------------------------------------------------------------
// Kernel 2: stage 12KB sorted tables into LDS with one TDM descriptor, then
// 4 consecutive queries per lane with b128 load/store.
// ---------------------------------------------------------------------------
__global__ __launch_bounds__(MAIN_THREADS)
void vq_assign(const float* __restrict__ ze,
               const float* __restrict__ wsBase,
               int* __restrict__ zOut,
               float* __restrict__ zqOut,
               int nElems) {
  // Single static LDS allocation -> starts at LDS offset 0 (TDM lds_addr = 0).
  __shared__ __attribute__((aligned(16))) unsigned char smem[TBL_ELEMS * 4];
  float*    sV = (float*)smem;                           // [3][512] sorted values
  unsigned* sI = (unsigned*)(smem + DDIMS * KCODES * 4); // [3][512] original indices

  // Wave 0 issues one tensor DMA: 3072 contiguous dwords ws -> LDS offset 0.
  if ((threadIdx.x >> 5) == 0) {
    unsigned long long ga = (unsigned long long)(uintptr_t)wsBase;
    // D# group 0 (ISA 08 §8.3): count=1, lds_addr=0, global_addr, type=2.
    v4u g0;
    g0.x = 1u;                                           // count = 1 valid descriptor
    g0.y = 0u;                                           // lds_addr = 0
    g0.z = (unsigned)(ga & 0xFFFFFFFFull);               // global_addr[31:0]
    g0.w = (unsigned)((ga >> 32) & 0x01FFFFFFull)        // global_addr[56:32]
         | 0x80000000u;                                  // type = 2 ("image")
    // D# group 1 (§8.4): mask=0 (no cluster), data_size=4B, 1-row tile of 3072.
    v8i g1;
    g1[0] = (int)(2u << 16);                             // data_size = 2 -> 4 bytes
    g1[1] = (int)(((unsigned)TBL_ELEMS & 0xFFFFu) << 16);// tensor_dim0[15:0] @63:48
    g1[2] = (int)(1u << 16);                             // tensor_dim0 hi=0; tensor_dim1=1
    g1[3] = (int)((unsigned)TBL_ELEMS << 16);            // tile_dim0 = 3072 @127:112
    g1[4] = 1;                                           // tile_dim1 = 1, tile_dim2 = 0
    g1[5] = (int)TBL_ELEMS;                              // tensor_dim0_stride lo32
    g1[6] = 0;                                           // stride hi16, dim1_stride lo16
    g1[7] = 0;
    v4i gz = {0, 0, 0, 0};
#if __clang_major__ >= 23
    v8i gz8 = {0, 0, 0, 0, 0, 0, 0, 0};
    __builtin_amdgcn_tensor_load_to_lds(g0, g1, gz, gz, gz8, 0);
#else
    __builtin_amdgcn_tensor_load_to_lds(g0, g1, gz, gz, 0);
#endif
    __builtin_amdgcn_s_wait_tensorcnt(0);
  }
  __syncthreads();

  const int base = (blockIdx.x * MAIN_THREADS + (int)threadIdx.x) * ITEMS;

  if (base + ITEMS <= nElems) {
    // Vector path: one global_load_b128, two b128 stores per lane.
    const float4 x4 = *(const float4*)(ze + base);
    const int d0 = base % DDIMS;
    const int d1 = (d0 + 1 == DDIMS) ? 0 : d0 + 1;
    const int d2 = (d1 + 1 == DDIMS) ? 0 : d1 + 1;
    const int d3 = d0;                                   // (d0 + 3) % 3

    int4   zr;
    float4 qr;
    vq_query(sV, sI, d0, x4.x, zr.x, qr.x);
    vq_query(sV, sI, d1, x4.y, zr.y, qr.y);
    vq_query(sV, sI, d2, x4.z, zr.z, qr.z);
    vq_query(sV, sI, d3, x4.w, zr.w, qr.w);

    *(int4*)(zOut + base)    = zr;                       // global_store_b128
    *(float4*)(zqOut + base) = qr;                       // global_store_b128
  } else {
    // Scalar tail (unused for B*D = 786432, kept for generality).
    for (int g = base; g < nElems; ++g) {
      float x = ze[g];
      int k; float vq;
      vq_query(sV, sI, g % DDIMS, x, k, vq);
      zOut[g]  = k;
      zqOut[g] = vq;
    }
  }
}

// ---------------------------------------------------------------------------
extern "C" void kernel_launch(void* const* d_in, const int* in_sizes, int n_in,
                              void* d_out, int out_size, void* d_ws, size_t ws_size,
                              hipStream_t stream) {
  const float* ze = (const float*)d_in[0];   // (B, D, 1) contiguous -> B*D floats
  const float* e  = (const float*)d_in[1];   // (K, D) row-major
  const int nElems = in_sizes[0];            // B*D = 786432

  float*    wsV = (float*)d_ws;                                  // 6 KB sorted values
  unsigned* wsI = (unsigned*)((char*)d_ws + DDIMS * KCODES * 4); // 6 KB indices

  int*   zOut  = (int*)d_out;                // z  as int32 bits, B*D elements
  float* zqOut = (float*)d_out + nElems;     // zq as float32,    B*D elements

  vq_sort_codebook<<<DDIMS, SORT_THREADS, 0, stream>>>(e, wsV, wsI);

  const int perBlock = MAIN_THREADS * ITEMS;
  const int blocks = (nElems + perBlock - 1) / perBlock;
  vq_assign<<<blocks, MAIN_THREADS, 0, stream>>>(ze, (const float*)d_ws,
                                                 zOut, zqOut, nElems);
}